// slotGATConv_52037823758816
// MI455X (gfx1250) — compile-verified
//
#include <hip/hip_runtime.h>
#include <hip/hip_bf16.h>

// ---- problem constants (match reference) ----
#define N_NODES   20000
#define N_EDGES   320000
#define NUM_NTYPE 4
#define IN_FEATS  64
#define OUT_FEATS 32
#define HEADS     4
#define EDGE_FEATS 64
#define NUM_ETYPES 8
#define SLOPE     0.2f
#define FDIM      (HEADS * NUM_NTYPE * OUT_FEATS)   // 512 floats per node in ft/rst

typedef __attribute__((ext_vector_type(2))) float v2f;
typedef __attribute__((ext_vector_type(8))) float v8f;

// ---------- helpers ----------
__device__ __forceinline__ void atomAddF(float* p, float v) {
  __hip_atomic_fetch_add(p, v, __ATOMIC_RELAXED, __HIP_MEMORY_SCOPE_AGENT);
}
// orderable self-inverse encoding of float bits into signed int (for atomic max)
__device__ __forceinline__ int encF(float f) {
  int i = __float_as_int(f);
  return i >= 0 ? i : (i ^ 0x7fffffff);
}
__device__ __forceinline__ float decF(int i) {
  return __int_as_float(i >= 0 ? i : (i ^ 0x7fffffff));
}

// ---------- init ----------
__global__ void zero_f32_kernel(float* __restrict__ p, int n) {
  int i = blockIdx.x * blockDim.x + threadIdx.x;
  if (i < n) p[i] = 0.0f;
}
__global__ void init_heads_kernel(float* __restrict__ ssum, int* __restrict__ m_enc, int n) {
  int i = blockIdx.x * blockDim.x + threadIdx.x;
  if (i < n) { ssum[i] = 0.0f; m_enc[i] = (int)0x80000000; }  // INT_MIN = enc(-inf-ish)
}

// ---------- per-etype attention bias table: ee[et][h] ----------
__global__ void ee_table_kernel(const float* __restrict__ edge_emb,  // (8,64)
                                const float* __restrict__ w_e,       // (64,256)
                                const float* __restrict__ attn_e,    // (1,4,64)
                                float* __restrict__ ee_table) {      // (8,4)
  int tid = threadIdx.x;
  if (tid >= NUM_ETYPES * HEADS) return;
  int et = tid >> 2, h = tid & 3;
  float acc = 0.0f;
  for (int f = 0; f < EDGE_FEATS; ++f) {
    float p = 0.0f;
    for (int k = 0; k < EDGE_FEATS; ++k)
      p += edge_emb[et * EDGE_FEATS + k] * w_e[k * (EDGE_FEATS * HEADS) + h * EDGE_FEATS + f];
    acc += p * attn_e[h * EDGE_FEATS + f];
  }
  ee_table[et * HEADS + h] = acc;
}

// ---------- WMMA GEMM: ft[n, h*128 + t*32 + o] = feat[n, t*64:].dot(fc[t][:, h*32+o]) ----------
// grid: 1250 M-tiles * 4 groups; block = 256 threads = 8 waves; each wave = one (t, ncol-tile)
__global__ __launch_bounds__(256) void gemm_ft_kernel(const float* __restrict__ feat,  // (N,256)
                                                      const float* __restrict__ fc,    // (4,64,128)
                                                      float* __restrict__ ft) {        // (N,512)
  const int lane = threadIdx.x & 31;
  const int wave = threadIdx.x >> 5;      // 0..7
  const int mtile = blockIdx.x >> 2;      // 0..1249
  const int wgrp  = blockIdx.x & 3;       // 0..3
  const int w  = wgrp * 8 + wave;         // 0..31
  const int t  = w >> 3;                  // node type
  const int n0 = (w & 7) * 16;            // column tile base in [0,128)
  const int m0 = mtile * 16;

  const int half = lane >> 4;             // 0: K%4 in {0,1}; 1: K%4 in {2,3}
  const int lm   = lane & 15;

  const float* Abase = feat + (size_t)(m0 + lm) * (NUM_NTYPE * IN_FEATS) + t * IN_FEATS + 2 * half;
  const float* Bbase = fc + (size_t)t * IN_FEATS * 128 + (size_t)(2 * half) * 128 + n0 + lm;

  v8f c = {};
#pragma unroll
  for (int k0 = 0; k0 < IN_FEATS; k0 += 4) {
    v2f a = *(const v2f*)(Abase + k0);     // A[m=lm][k0+2*half +{0,1}]
    v2f b;
    b.x = Bbase[(size_t)k0 * 128];         // B[k0+2*half  ][n0+lm]
    b.y = Bbase[(size_t)(k0 + 1) * 128];   // B[k0+2*half+1][n0+lm]
    c = __builtin_amdgcn_wmma_f32_16x16x4_f32(false, a, false, b, (short)0, c, false, false);
  }

  const int col = n0 + lm;                 // column in per-type (H*OUT)=128 space
  const int h = col >> 5;                  // head
  const int o = col & 31;                  // out-feat within head
#pragma unroll
  for (int v = 0; v < 8; ++v) {
    int node = m0 + v + 8 * half;          // D row mapping: vgpr v -> M=v (+8 for hi lanes)
    ft[(size_t)node * FDIM + h * 128 + t * OUT_FEATS + o] = c[v];
  }
}

// ---------- el/er: dot(ft[n,h,:], attn_{l,r}[h,:]) ----------
__global__ void elr_kernel(const float* __restrict__ ft,
                           const float* __restrict__ attn_l, const float* __restrict__ attn_r,
                           float* __restrict__ el, float* __restrict__ er) {
  int i = blockIdx.x * blockDim.x + threadIdx.x;
  if (i >= N_NODES * HEADS) return;
  int n = i >> 2, h = i & 3;
  const float* f  = ft + (size_t)n * FDIM + h * 128;
  const float* al = attn_l + h * 128;
  const float* ar = attn_r + h * 128;
  float sl = 0.0f, sr = 0.0f;
#pragma unroll 4
  for (int j = 0; j < 128; ++j) { float v = f[j]; sl += v * al[j]; sr += v * ar[j]; }
  el[i] = sl; er[i] = sr;
}

// ---------- edge logits + segment max ----------
__global__ void logits_max_kernel(const int* __restrict__ e_feat, const int* __restrict__ src,
                                  const int* __restrict__ dst,
                                  const float* __restrict__ el, const float* __restrict__ er,
                                  const float* __restrict__ ee_table,
                                  float* __restrict__ logits, int* __restrict__ m_enc) {
  int i = blockIdx.x * blockDim.x + threadIdx.x;
  if (i >= N_EDGES * HEADS) return;
  int e = i >> 2, h = i & 3;
  int s = src[e], d = dst[e];
  float x = el[s * HEADS + h] + er[d * HEADS + h] + ee_table[e_feat[e] * HEADS + h];
  x = x > 0.0f ? x : SLOPE * x;            // leaky relu
  logits[i] = x;
  atomicMax(&m_enc[d * HEADS + h], encF(x));
}

// ---------- exp + segment sum (in-place logits -> ex) ----------
__global__ void exp_sum_kernel(const int* __restrict__ dst, const int* __restrict__ m_enc,
                               float* __restrict__ exbuf, float* __restrict__ ssum) {
  int i = blockIdx.x * blockDim.x + threadIdx.x;
  if (i >= N_EDGES * HEADS) return;
  int e = i >> 2, h = i & 3;
  int d = dst[e];
  float m  = decF(m_enc[d * HEADS + h]);   // every edge's dst has >=1 edge, so m is finite
  float ex = expf(exbuf[i] - m);
  exbuf[i] = ex;
  atomAddF(&ssum[d * HEADS + h], ex);
}

// ---------- message passing: rst[dst] += a * ft[src]; one wave32 per edge ----------
__global__ __launch_bounds__(256) void scatter_kernel(const int* __restrict__ src,
                                                      const int* __restrict__ dst,
                                                      const float* __restrict__ exbuf,
                                                      const float* __restrict__ ssum,
                                                      const float* __restrict__ ft,
                                                      float* __restrict__ rst) {
  int gw   = (blockIdx.x * blockDim.x + threadIdx.x) >> 5;   // edge id
  int lane = threadIdx.x & 31;
  if (gw >= N_EDGES) return;
  int s = src[gw], d = dst[gw];
  int h = lane >> 3;                                          // 16 floats/lane, 128 floats/head
  float a = exbuf[gw * HEADS + h] / (ssum[d * HEADS + h] + 1e-16f);
  const float4* fsrc = (const float4*)(ft + (size_t)s * FDIM + lane * 16);
  float* out = rst + (size_t)d * FDIM + lane * 16;
#pragma unroll
  for (int j = 0; j < 4; ++j) {
    float4 v = fsrc[j];
    atomAddF(out + j * 4 + 0, v.x * a);
    atomAddF(out + j * 4 + 1, v.y * a);
    atomAddF(out + j * 4 + 2, v.z * a);
    atomAddF(out + j * 4 + 3, v.w * a);
  }
}

extern "C" void kernel_launch(void* const* d_in, const int* in_sizes, int n_in,
                              void* d_out, int out_size, void* d_ws, size_t ws_size,
                              hipStream_t stream) {
  const float* feat     = (const float*)d_in[0];
  const float* fc       = (const float*)d_in[1];
  const float* attn_l   = (const float*)d_in[2];
  const float* attn_r   = (const float*)d_in[3];
  const float* edge_emb = (const float*)d_in[4];
  const float* w_e      = (const float*)d_in[5];
  const float* attn_e   = (const float*)d_in[6];
  const int*   e_feat   = (const int*)d_in[7];
  const int*   src      = (const int*)d_in[8];
  const int*   dst      = (const int*)d_in[9];
  float* rst = (float*)d_out;

  // workspace layout (floats): total ~11.84M floats (~47.4 MB)
  float* ws       = (float*)d_ws;
  float* ft       = ws;                                   // N*512 = 10,240,000
  float* el       = ft + (size_t)N_NODES * FDIM;          // 80,000
  float* er       = el + N_NODES * HEADS;                 // 80,000
  float* ee_table = er + N_NODES * HEADS;                 // 32
  float* ssum     = ee_table + NUM_ETYPES * HEADS;        // 80,000
  int*   m_enc    = (int*)(ssum + N_NODES * HEADS);       // 80,000
  float* exbuf    = (float*)(m_enc + N_NODES * HEADS);    // E*4 = 1,280,000

  const int NH = N_NODES * HEADS;    // 80,000
  const int EH = N_EDGES * HEADS;    // 1,280,000

  // 0) init output + softmax accumulators
  zero_f32_kernel<<<(out_size + 255) / 256, 256, 0, stream>>>(rst, out_size);
  init_heads_kernel<<<(NH + 255) / 256, 256, 0, stream>>>(ssum, m_enc, NH);

  // 1) per-etype attention bias table (collapses the E x 64 x 256 GEMM to 8x4)
  ee_table_kernel<<<1, 32, 0, stream>>>(edge_emb, w_e, attn_e, ee_table);

  // 2) WMMA fp32 GEMM: feat -> ft   (1250 M-tiles * 4 wave-groups)
  gemm_ft_kernel<<<1250 * 4, 256, 0, stream>>>(feat, fc, ft);

  // 3) attention logits' node terms
  elr_kernel<<<(NH + 255) / 256, 256, 0, stream>>>(ft, attn_l, attn_r, el, er);

  // 4) edge softmax: leaky-relu logits + segment max
  logits_max_kernel<<<(EH + 255) / 256, 256, 0, stream>>>(e_feat, src, dst, el, er,
                                                          ee_table, exbuf, m_enc);
  // 5) exp + segment sum
  exp_sum_kernel<<<(EH + 255) / 256, 256, 0, stream>>>(dst, m_enc, exbuf, ssum);

  // 6) weighted gather/scatter message passing (one wave per edge)
  scatter_kernel<<<(N_EDGES * 32 + 255) / 256, 256, 0, stream>>>(src, dst, exbuf, ssum, ft, rst);
}